// PRIN_LSTM_10033043603514
// MI455X (gfx1250) — compile-verified
//
#include <hip/hip_runtime.h>
#include <hip/hip_bf16.h>

// ---------------------------------------------------------------------------
// Problem constants (match reference)
// ---------------------------------------------------------------------------
#define Bsz 16
#define Tt  1024
#define DI  256
#define Hh  512
#define NH  4
#define HD  128
#define DOo 256
#define PRUNE_TH 0.01f
#define LSTM_NWG 32

typedef __attribute__((ext_vector_type(2))) float v2f;
typedef __attribute__((ext_vector_type(8))) float v8f;

__device__ __forceinline__ v8f zero8() {
    v8f z = {0.f,0.f,0.f,0.f,0.f,0.f,0.f,0.f};
    return z;
}

// D = A(16x4) * B(4x16) + C, fp32 WMMA (CDNA5 V_WMMA_F32_16X16X4_F32)
__device__ __forceinline__ v8f wmma4(v2f a, v2f b, v8f c) {
    return __builtin_amdgcn_wmma_f32_16x16x4_f32(
        /*neg_a=*/false, a, /*neg_b=*/false, b,
        /*c_mod=*/(short)0, c, /*reuse_a=*/false, /*reuse_b=*/false);
}

// A-matrix fragment: X is row-major [16][ld] (M x K).
// ISA layout: V0 lanes0-15 K=k0, lanes16-31 K=k0+2; V1: K=k0+1 / k0+3.
__device__ __forceinline__ v2f load_a_frag(const float* X, int ld, int k0, int lane) {
    int hlf = lane >> 4;
    int m   = lane & 15;
    const float* p = X + (size_t)m * ld + k0 + (hlf ? 2 : 0);
    v2f f; f.x = p[0]; f.y = p[1];
    return f;
}

// B-matrix fragment from weight stored as W[n][k] (row n = output column),
// i.e. B[k][n] = W[n][k] so Y = X @ W^T.
// Assumed striping: V0 lanes0-15 K=k0, lanes16-31 K=k0+1; V1: K=k0+2 / k0+3.
__device__ __forceinline__ v2f load_b_frag(const float* W, int ld, int k0, int lane) {
    int hlf = lane >> 4;
    int n   = lane & 15;
    const float* p = W + (size_t)n * ld + k0 + hlf;
    v2f f; f.x = p[0]; f.y = p[2];
    return f;
}

__device__ __forceinline__ float sigmoidf_(float v) { return 1.f / (1.f + expf(-v)); }

// ---------------------------------------------------------------------------
// init: zero grid-barrier counter + both h state buffers (ws is poisoned)
// ---------------------------------------------------------------------------
__global__ void init_kernel(int* counter, float* h_bufs) {
    int i = blockIdx.x * 256 + threadIdx.x;
    if (i == 0) *counter = 0;
    if (i < 2 * Bsz * Hh) h_bufs[i] = 0.f;
}

// ---------------------------------------------------------------------------
// Persistent DPLSTM. 32 workgroups x 160 threads (5 waves).
// WG w owns H columns [w*16, w*16+16). Waves 0..3 compute the i/f/g/o gate
// slices (K = 256(x) + 512(h)), wave 4 computes the adaptive gate (K=512).
// Grid-synchronized per timestep via atomic counter barrier.
// h state is DOUBLE-BUFFERED: step t reads h_bufs[t&1], writes h_bufs[~t&1],
// so a fast workgroup can never clobber the h snapshot a slow workgroup is
// still staging (the per-step barrier separates the two buffers' roles).
// ---------------------------------------------------------------------------
__global__ void __launch_bounds__(160) lstm_kernel(
    const float* __restrict__ x,
    const float* __restrict__ W_ih, const float* __restrict__ b_ih,
    const float* __restrict__ W_hh, const float* __restrict__ b_hh,
    const float* __restrict__ W_ag, const float* __restrict__ b_ag,
    float* __restrict__ h_bufs, float* __restrict__ hs, int* barrier_counter) {

    __shared__ float A_s[Bsz * 768];      // [16][768] = x_t(256) | h(512)  (48 KB)
    __shared__ float gacc[5][16][16];     // gate tiles
    __shared__ float c_s[16][16];         // cell state for owned columns

    const int tid  = threadIdx.x;
    const int lane = tid & 31;
    const int wv   = tid >> 5;            // 0..4
    const int wg   = blockIdx.x;          // owns cols [wg*16, wg*16+16)

    for (int i = tid; i < 256; i += 160) c_s[i >> 4][i & 15] = 0.f;

    for (int t = 0; t < Tt; ++t) {
        const float* h_rd = h_bufs + (size_t)(t & 1) * Bsz * Hh;
        float*       h_wr = h_bufs + (size_t)((t + 1) & 1) * Bsz * Hh;

        __syncthreads();
        // stage A = [x_t | h_prev]  (float4 / b128 staging)
        for (int i = tid; i < (Bsz * DI) / 4; i += 160) {
            int b = i >> 6, k4 = i & 63;                       // 64 float4 per row
            *(float4*)&A_s[b * 768 + k4 * 4] =
                *(const float4*)&x[((size_t)b * Tt + t) * DI + k4 * 4];
        }
        for (int i = tid; i < (Bsz * Hh) / 4; i += 160) {
            int b = i >> 7, k4 = i & 127;                      // 128 float4 per row
            *(float4*)&A_s[b * 768 + 256 + k4 * 4] =
                *(const float4*)&h_rd[b * Hh + k4 * 4];
        }
        // prefetch next timestep's x rows into cache while we compute
        if (t + 1 < Tt) {
            for (int i = tid * 16; i < Bsz * DI; i += 160 * 16) {
                int b = i >> 8, k = i & 255;
                __builtin_prefetch(&x[((size_t)b * Tt + (t + 1)) * DI + k], 0, 3);
            }
        }
        __syncthreads();

        v8f acc0 = zero8(), acc1 = zero8();
        if (wv < 4) {
            // gate wv: weight rows [wv*H + wg*16, +16) of W_ih(2048x256)/W_hh(2048x512)
            const size_t gbase = (size_t)wv * Hh + wg * 16;
            const float* Wi = W_ih + gbase * DI;
            const float* Wr = W_hh + gbase * Hh;
            for (int k0 = 0; k0 < 768; k0 += 8) {
                v2f a0 = load_a_frag(A_s, 768, k0,     lane);
                v2f a1 = load_a_frag(A_s, 768, k0 + 4, lane);
                v2f b0 = (k0     < 256) ? load_b_frag(Wi, DI, k0,         lane)
                                        : load_b_frag(Wr, Hh, k0 - 256,   lane);
                v2f b1 = (k0 + 4 < 256) ? load_b_frag(Wi, DI, k0 + 4,     lane)
                                        : load_b_frag(Wr, Hh, k0 + 4 - 256, lane);
                acc0 = wmma4(a0, b0, acc0);
                acc1 = wmma4(a1, b1, acc1);
            }
        } else {
            const float* Wa = W_ag + (size_t)(wg * 16) * Hh;
            for (int k0 = 0; k0 < Hh; k0 += 8) {
                v2f a0 = load_a_frag(A_s + 256, 768, k0,     lane);
                v2f a1 = load_a_frag(A_s + 256, 768, k0 + 4, lane);
                v2f b0 = load_b_frag(Wa, Hh, k0,     lane);
                v2f b1 = load_b_frag(Wa, Hh, k0 + 4, lane);
                acc0 = wmma4(a0, b0, acc0);
                acc1 = wmma4(a1, b1, acc1);
            }
        }
        v8f acc = acc0 + acc1;
        {
            int hlf = lane >> 4, n = lane & 15;
            #pragma unroll
            for (int r = 0; r < 8; ++r) gacc[wv][r + 8 * hlf][n] = acc[r];
        }
        __syncthreads();

        // elementwise cell update for owned 16 columns x 16 batch rows
        for (int e = tid; e < 256; e += 160) {
            int b = e >> 4, n = e & 15;
            int col = wg * 16 + n;
            float ig = gacc[0][b][n] + b_ih[col]            + b_hh[col];
            float fg = gacc[1][b][n] + b_ih[Hh + col]       + b_hh[Hh + col];
            float gg = gacc[2][b][n] + b_ih[2 * Hh + col]   + b_hh[2 * Hh + col];
            float og = gacc[3][b][n] + b_ih[3 * Hh + col]   + b_hh[3 * Hh + col];
            float ag = gacc[4][b][n] + b_ag[col];
            float iv = sigmoidf_(ig), fv = sigmoidf_(fg);
            float gv = tanhf(gg),     ov = sigmoidf_(og), av = sigmoidf_(ag);
            float c_old = c_s[b][n];
            float c_new = av * (fv * c_old) + (1.f - av) * (iv * gv);
            float h_new = ov * tanhf(c_new);
            c_s[b][n] = c_new;
            h_wr[b * Hh + col] = h_new;                        // unpruned recurrent state
            hs[((size_t)t * Bsz + b) * Hh + col] =             // pruned output
                (fabsf(h_new) > PRUNE_TH) ? h_new : 0.f;
        }

        // grid barrier (all WGs resident: only 32 workgroups)
        __syncthreads();
        __threadfence();
        if (tid == 0) {
            atomicAdd(barrier_counter, 1);
            const int target = LSTM_NWG * (t + 1);
            while (atomicAdd(barrier_counter, 0) < target)
                __builtin_amdgcn_s_sleep(1);
        }
        __syncthreads();
        __threadfence();
    }
}

// ---------------------------------------------------------------------------
// K/V projection: Y[16384,512] = hs @ W^T + b. blockIdx.z selects K vs V.
// WG tile 128(M) x 64(N), 8 waves each doing a 16x64 strip, LDS-staged panels.
// ---------------------------------------------------------------------------
__global__ void __launch_bounds__(256) kv_proj_kernel(
    const float* __restrict__ hs,
    const float* __restrict__ Wk, const float* __restrict__ bk, float* __restrict__ Kbuf,
    const float* __restrict__ Wv, const float* __restrict__ bv, float* __restrict__ Vbuf) {

    const float* W  = blockIdx.z ? Wv : Wk;
    const float* bb = blockIdx.z ? bv : bk;
    float*       Y  = blockIdx.z ? Vbuf : Kbuf;

    __shared__ float Xs[128 * 64];   // 32 KB
    __shared__ float Ws_[64 * 64];   // 16 KB

    const int tid  = threadIdx.x;
    const int lane = tid & 31;
    const int wv   = tid >> 5;                 // 0..7 -> M sub-tile
    const size_t m0 = (size_t)blockIdx.x * 128;
    const int    n0 = blockIdx.y * 64;

    v8f acc[4];
    #pragma unroll
    for (int j = 0; j < 4; ++j) acc[j] = zero8();

    for (int kb = 0; kb < Hh; kb += 64) {
        __syncthreads();
        // float4 staging of X panel [128][64] and W panel [64][64]
        for (int i = tid; i < (128 * 64) / 4; i += 256) {
            int r = i >> 4, c4 = i & 15;
            *(float4*)&Xs[r * 64 + c4 * 4] =
                *(const float4*)&hs[(m0 + r) * Hh + kb + c4 * 4];
        }
        for (int i = tid; i < (64 * 64) / 4; i += 256) {
            int r = i >> 4, c4 = i & 15;
            *(float4*)&Ws_[r * 64 + c4 * 4] =
                *(const float4*)&W[(size_t)(n0 + r) * Hh + kb + c4 * 4];
        }
        // prefetch next K panel of the streaming X operand
        if (kb + 64 < Hh) {
            for (int i = tid; i < 128; i += 256)
                __builtin_prefetch(&hs[(m0 + i) * Hh + kb + 64], 0, 3);
        }
        __syncthreads();

        const float* Xw = Xs + wv * 16 * 64;
        for (int k0 = 0; k0 < 64; k0 += 4) {
            v2f a = load_a_frag(Xw, 64, k0, lane);
            #pragma unroll
            for (int j = 0; j < 4; ++j) {
                v2f bf = load_b_frag(Ws_ + (j * 16) * 64, 64, k0, lane);
                acc[j] = wmma4(a, bf, acc[j]);
            }
        }
    }

    const int hlf = lane >> 4, n = lane & 15;
    const size_t rbase = m0 + wv * 16;
    #pragma unroll
    for (int j = 0; j < 4; ++j)
        #pragma unroll
        for (int r = 0; r < 8; ++r) {
            int row = r + 8 * hlf;
            int col = n0 + j * 16 + n;
            Y[(rbase + row) * Hh + col] = acc[j][r] + bb[col];
        }
}

// ---------------------------------------------------------------------------
// Generic M=16 projection: Y[16,N] = X[16,K] @ W^T + bias (W is [N,K]).
// mode 0: store.  mode 1: Y = add0 + add1 - clamp(tanh(y), -5, 5)
// (feedback/err/comb fusion). Each wave owns one 16-col tile; block = 8 waves.
// ---------------------------------------------------------------------------
__global__ void __launch_bounds__(256) proj16_kernel(
    const float* __restrict__ X, const float* __restrict__ W,
    const float* __restrict__ bias,
    const float* __restrict__ add0, const float* __restrict__ add1,
    float* __restrict__ Y, int K, int N, int mode) {

    __shared__ float Xs[16 * 512];   // 32 KB K-panel

    const int tid  = threadIdx.x;
    const int lane = tid & 31;
    const int wv   = tid >> 5;
    const int nt   = blockIdx.x * 8 + wv;      // N tile index
    const bool active = (nt * 16 < N);

    v8f acc0 = zero8(), acc1 = zero8();

    for (int kb = 0; kb < K; kb += 512) {
        const int kw = (K - kb < 512) ? (K - kb) : 512;
        const int kw4 = kw >> 2;
        __syncthreads();
        for (int i = tid; i < 16 * kw4; i += 256) {
            int r = i / kw4, c4 = i - r * kw4;
            *(float4*)&Xs[r * 512 + c4 * 4] =
                *(const float4*)&X[(size_t)r * K + kb + c4 * 4];
        }
        __syncthreads();
        if (active) {
            const float* Wn = W + (size_t)(nt * 16) * K;   // 16 rows of W
            for (int k0 = 0; k0 < kw; k0 += 8) {
                v2f a0 = load_a_frag(Xs, 512, k0,     lane);
                v2f a1 = load_a_frag(Xs, 512, k0 + 4, lane);
                v2f b0 = load_b_frag(Wn, K, kb + k0,     lane);
                v2f b1 = load_b_frag(Wn, K, kb + k0 + 4, lane);
                acc0 = wmma4(a0, b0, acc0);
                acc1 = wmma4(a1, b1, acc1);
            }
        }
    }
    if (!active) return;

    v8f acc = acc0 + acc1;
    const int hlf = lane >> 4, n = lane & 15;
    const int col = nt * 16 + n;
    #pragma unroll
    for (int r = 0; r < 8; ++r) {
        int row = r + 8 * hlf;
        float y = acc[r] + bias[col];
        if (mode == 1) {
            float fb = tanhf(y);
            fb = fminf(5.f, fmaxf(-5.f, fb));
            y = add0[(size_t)row * N + col] + add1[(size_t)row * N + col] - fb;
        }
        Y[(size_t)row * N + col] = y;
    }
}

// ---------------------------------------------------------------------------
// Last-row attention: one block per (b, head). scores -> softmax (* head
// importance) -> weighted sum over V. Small VALU work (16.8M MACs total).
// ---------------------------------------------------------------------------
__global__ void __launch_bounds__(256) attn_kernel(
    const float* __restrict__ Qlast, const float* __restrict__ Kbuf,
    const float* __restrict__ Vbuf, const float* __restrict__ head_scaling,
    float* __restrict__ att_out) {

    __shared__ float sc[Tt];
    __shared__ float red[256];
    __shared__ float qv[HD];

    const int b = blockIdx.x >> 2;
    const int h = blockIdx.x & 3;
    const int tid = threadIdx.x;

    if (tid < HD) qv[tid] = Qlast[b * Hh + h * HD + tid];
    __syncthreads();

    // scores[t] = (Q . K_t) / sqrt(HD)
    for (int t = tid; t < Tt; t += 256) {
        const float* kr = Kbuf + ((size_t)t * Bsz + b) * Hh + h * HD;
        float s = 0.f;
        for (int d = 0; d < HD; ++d) s += qv[d] * kr[d];
        sc[t] = s * 0.08838834764831845f;   // 1/sqrt(128)
    }
    __syncthreads();

    // max reduction
    float lm = -1e30f;
    for (int t = tid; t < Tt; t += 256) lm = fmaxf(lm, sc[t]);
    red[tid] = lm; __syncthreads();
    for (int s = 128; s > 0; s >>= 1) {
        if (tid < s) red[tid] = fmaxf(red[tid], red[tid + s]);
        __syncthreads();
    }
    const float mx = red[0];
    __syncthreads();

    // exp + sum
    float ls = 0.f;
    for (int t = tid; t < Tt; t += 256) {
        float e = expf(sc[t] - mx);
        sc[t] = e;
        ls += e;
    }
    red[tid] = ls; __syncthreads();
    for (int s = 128; s > 0; s >>= 1) {
        if (tid < s) red[tid] += red[tid + s];
        __syncthreads();
    }
    const float sum = red[0];
    __syncthreads();

    // head importance softmax(head_scaling)[h]
    float s0 = head_scaling[0], s1 = head_scaling[1];
    float s2 = head_scaling[2], s3 = head_scaling[3];
    float hmax = fmaxf(fmaxf(s0, s1), fmaxf(s2, s3));
    float e0 = expf(s0 - hmax), e1 = expf(s1 - hmax);
    float e2 = expf(s2 - hmax), e3 = expf(s3 - hmax);
    float hsel = (h == 0) ? e0 : (h == 1) ? e1 : (h == 2) ? e2 : e3;
    const float scale = (hsel / (e0 + e1 + e2 + e3)) / sum;

    // att[d] = sum_t attn[t] * V[t, b, h, d]
    const int d  = tid & 127;
    const int hp = tid >> 7;    // split t range across two thread groups
    float a = 0.f;
    for (int t = hp * 512; t < hp * 512 + 512; ++t)
        a += sc[t] * Vbuf[((size_t)t * Bsz + b) * Hh + h * HD + d];
    red[tid] = a; __syncthreads();
    if (tid < 128)
        att_out[b * Hh + h * HD + tid] = (red[tid] + red[tid + 128]) * scale;
}

// ---------------------------------------------------------------------------
// Host-side launcher. Inputs (setup_inputs order):
//  0 x  1 W_ih 2 b_ih 3 W_hh 4 b_hh 5 W_ag 6 b_ag 7 Wq 8 bq 9 Wk 10 bk
//  11 Wv 12 bv 13 Wo 14 bo 15 head_scaling 16 Wp 17 bp 18 Wf 19 bf 20 Wfc 21 bfc
// ---------------------------------------------------------------------------
extern "C" void kernel_launch(void* const* d_in, const int* in_sizes, int n_in,
                              void* d_out, int out_size, void* d_ws, size_t ws_size,
                              hipStream_t stream) {
    (void)in_sizes; (void)n_in; (void)out_size; (void)ws_size;

    const float* x      = (const float*)d_in[0];
    const float* W_ih   = (const float*)d_in[1];
    const float* b_ih   = (const float*)d_in[2];
    const float* W_hh   = (const float*)d_in[3];
    const float* b_hh   = (const float*)d_in[4];
    const float* W_ag   = (const float*)d_in[5];
    const float* b_ag   = (const float*)d_in[6];
    const float* Wq     = (const float*)d_in[7];
    const float* bq     = (const float*)d_in[8];
    const float* Wk     = (const float*)d_in[9];
    const float* bk     = (const float*)d_in[10];
    const float* Wv     = (const float*)d_in[11];
    const float* bv     = (const float*)d_in[12];
    const float* Wo     = (const float*)d_in[13];
    const float* bo     = (const float*)d_in[14];
    const float* hsc    = (const float*)d_in[15];
    const float* Wp     = (const float*)d_in[16];   // [3,512,512] == [1536,512]
    const float* bp     = (const float*)d_in[17];   // [1536]
    const float* Wf     = (const float*)d_in[18];   // [512,1536]
    const float* bf     = (const float*)d_in[19];
    const float* Wfc    = (const float*)d_in[20];   // [256,512]
    const float* bfc    = (const float*)d_in[21];
    float* out = (float*)d_out;                     // [16,256] fp32

    // workspace layout (floats)
    float* ws = (float*)d_ws;
    int*   counter = (int*)d_ws;                    // first 256 B reserved
    const size_t HCUR  = 64;                                       // h double buffer [2][16][512]
    const size_t HS    = HCUR + (size_t)2 * Bsz * Hh;              // pruned hs [T*B, H]
    const size_t KB    = HS   + (size_t)Tt * Bsz * Hh;
    const size_t VB    = KB   + (size_t)Tt * Bsz * Hh;
    const size_t QL    = VB   + (size_t)Tt * Bsz * Hh;
    const size_t ATT   = QL   + (size_t)Bsz * Hh;
    const size_t ATTR  = ATT  + (size_t)Bsz * Hh;
    const size_t COMB1 = ATTR + (size_t)Bsz * Hh;                  // combined [16,1536]
    const size_t COMB  = COMB1 + (size_t)Bsz * 3 * Hh;             // comb_last [16,512]

    float* h_bufs  = ws + HCUR;
    float* hs      = ws + HS;
    float* Kbuf    = ws + KB;
    float* Vbuf    = ws + VB;
    float* Qlast   = ws + QL;
    float* att     = ws + ATT;
    float* attrow  = ws + ATTR;
    float* comb1   = ws + COMB1;
    float* combl   = ws + COMB;
    const float* out_last = hs + (size_t)(Tt - 1) * Bsz * Hh;      // pruned h at t=T-1

    // 1) init barrier + h state (both buffers)
    init_kernel<<<64, 256, 0, stream>>>(counter, h_bufs);

    // 2) persistent LSTM over T steps (grid-synced, double-buffered h)
    lstm_kernel<<<LSTM_NWG, 160, 0, stream>>>(x, W_ih, b_ih, W_hh, b_hh,
                                              W_ag, b_ag, h_bufs, hs, counter);

    // 3) K and V projections (z selects)
    kv_proj_kernel<<<dim3((Tt * Bsz) / 128, Hh / 64, 2), 256, 0, stream>>>(
        hs, Wk, bk, Kbuf, Wv, bv, Vbuf);

    // 4) Q at last timestep only: [16,512] = out_last @ Wq^T + bq
    proj16_kernel<<<(Hh + 127) / 128, 256, 0, stream>>>(
        out_last, Wq, bq, nullptr, nullptr, Qlast, Hh, Hh, 0);

    // 5) last-row attention per (b, head)
    attn_kernel<<<Bsz * NH, 256, 0, stream>>>(Qlast, Kbuf, Vbuf, hsc, att);

    // 6) output projection of attention: attrow = att @ Wo^T + bo
    proj16_kernel<<<(Hh + 127) / 128, 256, 0, stream>>>(
        att, Wo, bo, nullptr, nullptr, attrow, Hh, Hh, 0);

    // 7) predictive coding: combined[16,1536] = attrow @ Wp_flat^T + bp
    proj16_kernel<<<(3 * Hh + 127) / 128, 256, 0, stream>>>(
        attrow, Wp, bp, nullptr, nullptr, comb1, Hh, 3 * Hh, 0);

    // 8) comb = out_last + attrow - clamp(tanh(combined @ Wf^T + bf), +-5)
    proj16_kernel<<<(Hh + 127) / 128, 256, 0, stream>>>(
        comb1, Wf, bf, out_last, attrow, combl, 3 * Hh, Hh, 1);

    // 9) final head: out[16,256] = comb @ Wfc^T + bfc
    proj16_kernel<<<(DOo + 127) / 128, 256, 0, stream>>>(
        combl, Wfc, bfc, nullptr, nullptr, out, Hh, DOo, 0);
}